// TanBayesNetClassifierSubsetStructurePenalty_37194416783934
// MI455X (gfx1250) — compile-verified
//
#include <hip/hip_runtime.h>
#include <hip/hip_bf16.h>

#define B_   8192
#define F_   32
#define V_   64
#define C_   32
#define K_   8
#define KP_  7
#define NF   (F_ - 1)        // 31
#define NFK  (NF * KP_)      // 217
#define EPS_ 1e-6f

typedef __attribute__((ext_vector_type(2))) float v2f;
typedef __attribute__((ext_vector_type(8))) float v8f;

// workspace float layout
#define WS_W0   0                    // 31  : w[f,0]
#define WS_WP   31                   // 217 : w[f,k+1] (0 for invalid slots)
#define WS_BASE 248                  // 32  : cl - lse0 - sum_f w0*lse_self
#define WS_LSEW 288                  // NFK*64*32 = 444416 : w[f,k+1]*lse_pair[f,k,xc,c]
#define WS_U    (WS_LSEW + NFK*V_*C_)   // 32*64*32 = 65536 : folded per-column table
// total = 510240 floats ~ 1.95 MB

// ---------------------------------------------------------------------------
// Kernel 1: gumbel-softmax gates + per-class base constant (1 block, tiny)
// ---------------------------------------------------------------------------
__global__ void k_setup(const float* __restrict__ class_logits,
                        const float* __restrict__ table0,
                        const float* __restrict__ self_tables,
                        const float* __restrict__ structure_logits,
                        const float* __restrict__ u_noise,
                        float* __restrict__ ws) {
  const float NEG_INF = -__builtin_inff();
  int t = threadIdx.x;
  if (t < NF) {
    int f = t;
    int kmax = min(f + 1, KP_);            // # valid pair slots for this row
    float lv[K_];
    float m = NEG_INF;
    for (int k = 0; k < K_; ++k) {
      bool valid = (k == 0) || ((k - 1) < kmax);
      if (valid) {
        float u = u_noise[f * K_ + k];
        float g = -__logf(-__logf(u + EPS_) + EPS_);
        float v = structure_logits[f * K_ + k] + g;   // TEMP = 1
        lv[k] = v;
        m = fmaxf(m, v);
      } else {
        lv[k] = NEG_INF;
      }
    }
    float s = 0.f, e[K_];
    for (int k = 0; k < K_; ++k) {
      e[k] = (lv[k] == NEG_INF) ? 0.f : __expf(lv[k] - m);
      s += e[k];
    }
    float inv = 1.f / s;
    ws[WS_W0 + f] = e[0] * inv;
    for (int k = 0; k < KP_; ++k) ws[WS_WP + f * KP_ + k] = e[k + 1] * inv;
  }
  __syncthreads();
  if (t < C_) {
    int c = t;
    // logsumexp(class_logits)
    float m = NEG_INF, s = 0.f;
    for (int i = 0; i < C_; ++i) {
      float v = class_logits[i];
      if (v > m) { s = s * __expf(m - v) + 1.f; m = v; } else s += __expf(v - m);
    }
    float cl = class_logits[c] - (m + __logf(s));
    // logsumexp over V of table0[:,c]
    m = NEG_INF; s = 0.f;
    for (int v = 0; v < V_; ++v) {
      float x = table0[v * C_ + c];
      if (x > m) { s = s * __expf(m - x) + 1.f; m = x; } else s += __expf(x - m);
    }
    float lse0 = m + __logf(s);
    // sum_f w0[f] * logsumexp_v(self_tables[f,:,c])  (batch-independent)
    float Cs = 0.f;
    for (int f = 0; f < NF; ++f) {
      m = NEG_INF; s = 0.f;
      for (int v = 0; v < V_; ++v) {
        float x = self_tables[(f * V_ + v) * C_ + c];
        if (x > m) { s = s * __expf(m - x) + 1.f; m = x; } else s += __expf(x - m);
      }
      Cs += ws[WS_W0 + f] * (m + __logf(s));
    }
    ws[WS_BASE + c] = cl - lse0 - Cs;
  }
}

// ---------------------------------------------------------------------------
// Kernel 2: lsew[f,k,xc,c] = w[f,k+1] * (m + log Σ_xs exp(pt - m))
// Σ_xs exp via V_WMMA_F32_16X16X4_F32 with A = ones (exact column-sum:
// D[m,n] = Σ_k B[k,n]; any hw slot→k permutation is absorbed by the sum).
// One wave per (fk, xc); streams the 111MB table once (+hot re-read).
// ---------------------------------------------------------------------------
__global__ void k_pair_lse(const float* __restrict__ pair_tables,
                           float* __restrict__ ws) {
  const int fk   = blockIdx.x >> 4;                     // 0..216
  const int wave = threadIdx.x >> 5;                    // 0..3
  const int lane = threadIdx.x & 31;
  const int xc   = ((blockIdx.x & 15) << 2) + wave;     // 0..63
  const int f    = fk / KP_;
  const int k    = fk - f * KP_;
  if (k >= min(f + 1, KP_)) return;                     // structurally invalid slot
  const float wpv = ws[WS_WP + fk];
  const float* pt = pair_tables + (size_t)fk * (V_ * V_ * C_);

  // pass 1: per-class running max over xs (lane = c, coalesced 128B rows)
  float m = -__builtin_inff();
  for (int xs = 0; xs < V_; ++xs)
    m = fmaxf(m, pt[(xs * V_ + xc) * C_ + lane]);

  const int c16 = lane & 15;
  const int hi  = lane >> 4;
  float* lsew = ws + WS_LSEW;
  for (int cb = 0; cb < C_; cb += 16) {
    float mm = __shfl(m, cb + c16, 32);                 // max for this column
    v8f acc = {};
    const v2f ones = {1.f, 1.f};
    for (int j = 0; j < 16; ++j) {                      // 64 xs in chunks of 4
      int xs0 = 4 * j + 2 * hi;
      float b0 = __expf(pt[( xs0      * V_ + xc) * C_ + cb + c16] - mm);
      float b1 = __expf(pt[((xs0 + 1) * V_ + xc) * C_ + cb + c16] - mm);
      v2f bb = {b0, b1};
      acc = __builtin_amdgcn_wmma_f32_16x16x4_f32(false, ones, false, bb,
                                                  (short)0, acc, false, false);
    }
    // D rows identical; VGPR0 lane l = colsum[l&15]
    float lse = mm + __logf(acc[0]);
    if (lane < 16)
      lsew[((size_t)fk * V_ + xc) * C_ + cb + lane] = wpv * lse;
  }
}

// ---------------------------------------------------------------------------
// Kernel 2b: fold per-x-column table
//   U[0 ,v,c] = table0[v,c]                       - Σ_k lsew[(k,k),v,c]
//   U[j ,v,c] = w0[j-1]*self_tables[j-1,v,c]      - Σ_k lsew[(j+k,k),v,c]  (1..30)
//   U[31,v,c] = w0[30]*self_tables[30,v,c]
// This turns 196 lse-gathers per batch row into 0 (absorbed into the 32
// per-column gathers). One wave per (j,v): 2048 waves.
// ---------------------------------------------------------------------------
__global__ void k_fold(const float* __restrict__ table0,
                       const float* __restrict__ self_tables,
                       float* __restrict__ ws) {
  const int wave = threadIdx.x >> 5;
  const int lane = threadIdx.x & 31;                    // c
  const int jv   = blockIdx.x * (blockDim.x >> 5) + wave;   // 0..2047
  const int j    = jv >> 6;                             // x column 0..31
  const int v    = jv & 63;
  const float* lsew = ws + WS_LSEW;

  float acc;
  if (j == 0) {
    acc = table0[v * C_ + lane];
  } else {
    acc = ws[WS_W0 + (j - 1)] * self_tables[(((j - 1) * V_) + v) * C_ + lane];
  }
  if (j <= 30) {
    const int kcnt = min(6, 30 - j) + 1;                // valid (f,k)=(j+k,k)
    for (int k = 0; k < kcnt; ++k) {
      const int fk = (j + k) * KP_ + k;
      acc -= lsew[((size_t)fk * V_ + v) * C_ + lane];
    }
  }
  ws[WS_U + ((size_t)j * V_ + v) * C_ + lane] = acc;
}

// ---------------------------------------------------------------------------
// Kernel 3: main gather-accumulate. One wave per batch row, lane = class.
// Exactly one coalesced 128B gather per model term: 32 U rows + 196 pair rows.
// ---------------------------------------------------------------------------
__global__ void k_main(const int* __restrict__ x,
                       const float* __restrict__ pair_tables,
                       const float* __restrict__ ws,
                       float* __restrict__ out) {
  __shared__ float sm[WS_BASE + C_];                    // gates + base: 1.1KB
  for (int i = threadIdx.x; i < WS_BASE + C_; i += blockDim.x) sm[i] = ws[i];
  __syncthreads();

  const int lane = threadIdx.x & 31;
  const int wave = threadIdx.x >> 5;
  const int b    = blockIdx.x * (blockDim.x >> 5) + wave;
  const int xi   = x[b * F_ + lane];                    // whole x row in the wave
  const float* U = ws + WS_U;

  float acc = sm[WS_BASE + lane];

  // per-column folded tables (table0, scaled self, minus folded pair-LSE)
  for (int j = 0; j < F_; ++j) {
    int vj = __shfl(xi, j, 32);
    acc += U[((size_t)j * V_ + vj) * C_ + lane];
  }

  // raw pair gathers, scaled by gate weights
  for (int f = 0; f < NF; ++f) {
    const int kmax = min(f + 1, KP_);                   // exact w==0 structure
    const int xsv  = __shfl(xi, f + 1, 32);
    for (int k = 0; k < kmax; ++k) {
      const int fk  = f * KP_ + k;
      const int xcv = __shfl(xi, f - k, 32);            // cond_index[f,k] = f-k
      if (k + 1 < kmax) {                               // prefetch next pair row
        int xcn = __shfl(xi, f - k - 1, 32);
        __builtin_prefetch(
            pair_tables + (((size_t)(fk + 1) * V_ + xsv) * V_ + xcn) * C_ + lane,
            0, 0);
      }
      acc += sm[WS_WP + fk] *
             pair_tables[(((size_t)fk * V_ + xsv) * V_ + xcv) * C_ + lane];
    }
  }
  out[b * C_ + lane] = acc;
}

// ---------------------------------------------------------------------------
extern "C" void kernel_launch(void* const* d_in, const int* in_sizes, int n_in,
                              void* d_out, int out_size, void* d_ws, size_t ws_size,
                              hipStream_t stream) {
  const int*   x   = (const int*)  d_in[0];
  // d_in[1] = training (unused by reference)
  const float* cls = (const float*)d_in[2];
  const float* t0  = (const float*)d_in[3];
  const float* st  = (const float*)d_in[4];
  const float* pt  = (const float*)d_in[5];
  const float* sl  = (const float*)d_in[6];
  const float* un  = (const float*)d_in[7];
  // d_in[8]=cond_index, d_in[9]=valid_mask: deterministic structure, rebuilt in-kernel
  float* ws  = (float*)d_ws;   // needs ~1.95 MB
  float* out = (float*)d_out;

  hipLaunchKernelGGL(k_setup,    dim3(1),        dim3(64),  0, stream,
                     cls, t0, st, sl, un, ws);
  hipLaunchKernelGGL(k_pair_lse, dim3(NFK * 16), dim3(128), 0, stream, pt, ws);
  hipLaunchKernelGGL(k_fold,     dim3(256),      dim3(256), 0, stream, t0, st, ws);
  hipLaunchKernelGGL(k_main,     dim3(B_ / 8),   dim3(256), 0, stream,
                     x, pt, ws, out);
}